// Simple_50740743635546
// MI455X (gfx1250) — compile-verified
//
#include <hip/hip_runtime.h>
#include <math.h>

// ============================================================================
// ACT recurrence on MI455X (gfx1250).
//
// Per-step cost = two 4096x4096 f32 matvecs = ~128MB weight traffic, ~67 MFLOP
// -> pure bandwidth problem (0.5 FLOP/B). The per-step weight working set
// (128MB) fits in the 192MB L2, so steps >0 run at L2 bandwidth. Matvecs are
// executed with V_WMMA_F32_16X16X4_F32 (exact f32, matches reference math):
//
//   * A tile (16x4 f32, documented layout): lane(0..15)=row M, VGPR0 holds
//     K=0 (lanes 0-15) / K=2 (lanes 16-31), VGPR1 holds K=1 / K=3.
//   * B (4x16) = state vector broadcast across all 16 N columns. Under
//     broadcast, only B's K->(VGPR,lane-half) mapping matters; we assume it
//     mirrors A's (VGPR0: K0 lo / K2 hi, VGPR1: K1 lo / K3 hi), which a single
//     8B-aligned ds_load_b64 at offset (k0 + 2*half)*4 produces directly.
//   * D is then replicated across N: lane 0's 8 acc VGPRs are rows R0..R0+7,
//     lane 16's are rows R0+8..R0+15. No cross-lane reduction needed.
//
// 256 waves (128 blocks x 2 waves) cover the 256 row-tiles; each wave streams
// 16 rows x 16KB. EXEC is all-ones on every WMMA path (uniform branches only
// before the loop), as the ISA requires.
// ============================================================================

typedef float v2f __attribute__((ext_vector_type(2)));
typedef float v8f __attribute__((ext_vector_type(8)));

#define H 4096
#define LDH 8193          // W_hidden row stride (1 + IN + H)
#define MAX_STEPS 128
#define ONE_MINUS_EPS 0.99f

// y[R0..R0+15] = W(rows from Wrow, row-major stride ld) @ vec(LDS, length K).
// Returns the WMMA accumulator: lane 0 holds rows R0..R0+7 in acc[0..7],
// lane 16 holds rows R0+8..R0+15.
__device__ __forceinline__ v8f wmma_matvec16(const float* __restrict__ Wrow,
                                             int ld,
                                             const float* __restrict__ lds_vec,
                                             int K) {
  const int lane = threadIdx.x & 31;
  const int hi   = lane >> 4;     // lane-half: selects K=+2 for A/B layout
  const int m    = lane & 15;     // row within the 16-row tile
  const float* aptr = Wrow + (size_t)m * (size_t)ld + 2 * hi;
  const float* bptr = lds_vec + 2 * hi;
  v8f acc = {0.f, 0.f, 0.f, 0.f, 0.f, 0.f, 0.f, 0.f};
#pragma unroll 8
  for (int k0 = 0; k0 < K; k0 += 4) {
    v2f a, b;
    a.x = aptr[k0];       // A: K = k0 + 2*hi
    a.y = aptr[k0 + 1];   // A: K = k0 + 2*hi + 1
    b.x = bptr[k0];       // B broadcast across N, same K mapping (ds_load_b64)
    b.y = bptr[k0 + 1];
    acc = __builtin_amdgcn_wmma_f32_16x16x4_f32(false, a, false, b,
                                                (short)0, acc, false, false);
  }
  return acc;
}

__device__ __forceinline__ void preload_vec(float* lds, const float* g, int n) {
  const float4* g4 = (const float4*)g;
  float4* l4 = (float4*)lds;
  for (int i = threadIdx.x; i < n / 4; i += blockDim.x) l4[i] = g4[i];
  __syncthreads();
}

// --- init: zero outputs, copy s into state buffer 0, zero scalars -----------
__global__ void act_init(float* __restrict__ out, float* __restrict__ state0,
                         const float* __restrict__ s, float* __restrict__ scal) {
  int i = blockIdx.x * blockDim.x + threadIdx.x;
  if (i < H + H + 1) out[i] = 0.f;                 // acc_out, acc_state, ponder
  if (i < H)         state0[i] = s[i];
  if (i == 0) { scal[0] = 0.f; scal[1] = 0.f; scal[2] = 0.f; } // cum, done, w
}

// --- hoisted constant: c = W_hidden[:,1:4097] @ x + b_hidden ----------------
__global__ void act_precompute_c(const float* __restrict__ Whid,
                                 const float* __restrict__ x,
                                 const float* __restrict__ bh,
                                 float* __restrict__ cvec) {
  __shared__ float lvec[H];
  preload_vec(lvec, x, H);
  const int wave = threadIdx.x >> 5;
  const int wid  = blockIdx.x * (blockDim.x >> 5) + wave;
  const int R0   = wid * 16;
  v8f acc = wmma_matvec16(Whid + (size_t)R0 * LDH + 1, LDH, lvec, H);
  const int lane = threadIdx.x & 31;
  if ((lane & 15) == 0) {
    const int rb = R0 + (lane >> 4) * 8;
#pragma unroll
    for (int j = 0; j < 8; ++j) cvec[rb + j] = acc[j] + bh[rb + j];
  }
}

// --- per step: new_state = Wh_state @ state + c (+flag*Wh[:,0]); halt partials
__global__ void act_step_state(const float* __restrict__ Whid,
                               const float* __restrict__ Whalt,
                               const float* __restrict__ cvec,
                               const float* __restrict__ sin,
                               float* __restrict__ sout,
                               float* __restrict__ halt_partials,
                               const float* __restrict__ scal,
                               float flag) {
  if (scal[1] != 0.f) return;                      // done: everything is a NOP
  __shared__ float lvec[H];
  preload_vec(lvec, sin, H);
  const int wave = threadIdx.x >> 5;
  const int wid  = blockIdx.x * (blockDim.x >> 5) + wave;
  const int R0   = wid * 16;
  v8f acc = wmma_matvec16(Whid + (size_t)R0 * LDH + 4097, LDH, lvec, H);
  const int lane = threadIdx.x & 31;
  const int hi   = lane >> 4;
  if ((lane & 15) == 0) {                          // lanes 0 and 16 own results
    const int rb = R0 + hi * 8;
    float hp = 0.f;
#pragma unroll
    for (int j = 0; j < 8; ++j) {
      const int r = rb + j;
      float ns = acc[j] + cvec[r] + flag * Whid[(size_t)r * LDH]; // z[0]=flag
      sout[r] = ns;
      hp += Whalt[r] * ns;                         // halt-logit partial
    }
    halt_partials[wid * 2 + hi] = hp;
  }
}

// --- per step: reduce halt logit, compute p, w, cum, done, ponder -----------
__global__ void act_finalize(float* __restrict__ scal,
                             const float* __restrict__ partials,
                             const float* __restrict__ bhalt,
                             float* __restrict__ out, int step) {
  __shared__ float red[256];
  const int t = threadIdx.x;
  red[t] = partials[t] + partials[t + 256];        // 512 wave-partials
  __syncthreads();
  for (int s2 = 128; s2 > 0; s2 >>= 1) {
    if (t < s2) red[t] += red[t + s2];
    __syncthreads();
  }
  if (t == 0) {
    if (scal[1] != 0.f) { scal[2] = 0.f; return; } // already halted: w = 0
    const float logit = red[0] + bhalt[0];
    const float p = 1.f / (1.f + expf(-logit));
    const float cum = scal[0];
    const bool will_halt = (cum + p >= ONE_MINUS_EPS) || (step == MAX_STEPS - 1);
    scal[2] = will_halt ? (1.f - cum) : p;         // w (>0 always when active)
    if (will_halt) {
      out[2 * H] = (float)step + 1.f + (1.f - cum); // ponder
      scal[1] = 1.f;                                // done
    }
    scal[0] = cum + p;
  }
}

// --- per step: acc_out += w*(W_out@ns + b_out); acc_state += w*ns -----------
__global__ void act_step_out(const float* __restrict__ Wout,
                             const float* __restrict__ bout,
                             const float* __restrict__ ns,
                             float* __restrict__ out,
                             const float* __restrict__ scal) {
  const float w = scal[2];
  if (w == 0.f) return;                            // inactive step
  __shared__ float lvec[H];
  preload_vec(lvec, ns, H);
  const int wave = threadIdx.x >> 5;
  const int wid  = blockIdx.x * (blockDim.x >> 5) + wave;
  const int R0   = wid * 16;
  v8f acc = wmma_matvec16(Wout + (size_t)R0 * H, H, lvec, H);
  const int lane = threadIdx.x & 31;
  if ((lane & 15) == 0) {
    const int rb = R0 + (lane >> 4) * 8;
#pragma unroll
    for (int j = 0; j < 8; ++j) {
      const int r = rb + j;
      out[r]     += w * (acc[j] + bout[r]);        // acc_out
      out[H + r] += w * lvec[r];                   // acc_state
    }
  }
}

extern "C" void kernel_launch(void* const* d_in, const int* in_sizes, int n_in,
                              void* d_out, int out_size, void* d_ws, size_t ws_size,
                              hipStream_t stream) {
  const float* x     = (const float*)d_in[0];
  const float* s     = (const float*)d_in[1];
  const float* Whid  = (const float*)d_in[2];   // [4096 x 8193]
  const float* bh    = (const float*)d_in[3];
  const float* Whalt = (const float*)d_in[4];   // [4096]
  const float* bhalt = (const float*)d_in[5];   // [1]
  const float* Wout  = (const float*)d_in[6];   // [4096 x 4096]
  const float* bout  = (const float*)d_in[7];
  float* out = (float*)d_out;                   // [4096 out | 4096 hid | 1 ponder]

  float* ws       = (float*)d_ws;               // ~51KB scratch layout:
  float* cvec     = ws;                         // [4096] hoisted x-projection
  float* buf0     = ws + 4096;                  // [4096] state ping
  float* buf1     = ws + 8192;                  // [4096] state pong
  float* partials = ws + 12288;                 // [512]  halt partials
  float* scal     = ws + 12800;                 // [3]    cum, done, w

  act_init<<<(2 * H + 256) / 256, 256, 0, stream>>>(out, buf0, s, scal);
  act_precompute_c<<<128, 64, 0, stream>>>(Whid, x, bh, cvec);

  for (int i = 0; i < MAX_STEPS; ++i) {
    float* sin  = (i & 1) ? buf1 : buf0;
    float* sout = (i & 1) ? buf0 : buf1;
    act_step_state<<<128, 64, 0, stream>>>(Whid, Whalt, cvec, sin, sout,
                                           partials, scal, i == 0 ? 1.f : 0.f);
    act_finalize<<<1, 256, 0, stream>>>(scal, partials, bhalt, out, i);
    act_step_out<<<128, 64, 0, stream>>>(Wout, bout, sout, out, scal);
  }
}